// GraphAttentionModel_59519656788179
// MI455X (gfx1250) — compile-verified
//
#include <hip/hip_runtime.h>
#include <math.h>

#define DIN 128
#define H1 8
#define C1 16
#define F1 128   /* H1*C1 */
#define DOUT 32
#define NEG_SLOPE 0.2f

typedef float v2f __attribute__((ext_vector_type(2)));
typedef float v8f __attribute__((ext_vector_type(8)));

// ---------------------------------------------------------------------------
// Generic fp32 WMMA GEMM:  C[M x NC] = A[M x K] * B[K x NC]
// One wave32 computes one 16x16 tile via V_WMMA_F32_16X16X4_F32.
// Requires M % 16 == 0, NC % 16 == 0, K % 4 == 0 (true for all calls here).
// blockDim.x == 256 (8 waves per block); tile selection is wave-uniform so
// EXEC is all-ones at every WMMA (ISA requirement).
// ---------------------------------------------------------------------------
__global__ void gemm_wmma_f32(const float* __restrict__ A,
                              const float* __restrict__ B,
                              float* __restrict__ C,
                              int M, int K, int NC) {
    int wave   = blockIdx.x * (blockDim.x >> 5) + (threadIdx.x >> 5);
    int tilesN = NC >> 4;
    int tilesM = M >> 4;
    if (wave >= tilesM * tilesN) return;          // wave-uniform
    int mt   = wave / tilesN;
    int nt   = wave % tilesN;
    int lane = threadIdx.x & 31;
    int half = lane >> 4;                         // 0: lanes 0-15, 1: lanes 16-31
    int l    = lane & 15;

    v8f c = {};
    const float* arow = A + (size_t)(mt * 16 + l) * K;   // A row for M = mt*16 + l
    const float* bcol = B + (size_t)(nt * 16 + l);       // B column base, N = nt*16 + l
    for (int k0 = 0; k0 < K; k0 += 4) {
        // 32-bit A 16x4 layout: lanes 0-15 -> K = k0+{0,1}; lanes 16-31 -> K = k0+{2,3}
        v2f a, b;
        a.x = arow[k0 + 2 * half + 0];
        a.y = arow[k0 + 2 * half + 1];
        // 32-bit B 4x16 layout (mirror of A): row K striped per VGPR, lane-half adds 2
        b.x = bcol[(size_t)(k0 + 2 * half + 0) * NC];
        b.y = bcol[(size_t)(k0 + 2 * half + 1) * NC];
        c = __builtin_amdgcn_wmma_f32_16x16x4_f32(false, a, false, b,
                                                  (short)0, c, false, false);
    }
    // 32-bit C/D 16x16 layout: VGPR v -> M = v + 8*half, N = l
    #pragma unroll
    for (int v = 0; v < 8; ++v) {
        int row = mt * 16 + v + 8 * half;
        C[(size_t)row * NC + nt * 16 + l] = c[v];
    }
}

// ---------------------------------------------------------------------------
// Monotone float <-> uint mapping so float max == unsigned max
// ---------------------------------------------------------------------------
__device__ __forceinline__ unsigned fenc(float f) {
    unsigned u = __float_as_uint(f);
    return (u & 0x80000000u) ? ~u : (u | 0x80000000u);
}
__device__ __forceinline__ float fdec(unsigned u) {
    return __uint_as_float((u & 0x80000000u) ? (u ^ 0x80000000u) : ~u);
}

__global__ void zero_f32(float* __restrict__ p, long long n) {
    long long t = blockIdx.x * (long long)blockDim.x + threadIdx.x;
    if (t < n) p[t] = 0.0f;
}

// ---------------------------------------------------------------------------
// Edge pass 1: per (edge, head) attention logit -> atomicMax into maxbuf[dst,h]
// Edges e >= E are self loops (src = dst = e - E).
// ---------------------------------------------------------------------------
__global__ void edge_max(const float* __restrict__ xl, const float* __restrict__ xr,
                         const float* __restrict__ att, const int* __restrict__ ei,
                         int E, int N, int heads, int C,
                         unsigned* __restrict__ maxbuf) {
    long long tid   = blockIdx.x * (long long)blockDim.x + threadIdx.x;
    long long total = (long long)(E + N) * heads;
    if (tid >= total) return;
    int h = (int)(tid % heads);
    int e = (int)(tid / heads);
    int src, dst;
    if (e < E) { src = ei[e]; dst = ei[E + e]; } else { src = dst = e - E; }
    int F = heads * C;
    const float* pl = xl + (size_t)src * F + h * C;
    const float* pr = xr + (size_t)dst * F + h * C;
    const float* pa = att + h * C;
    float s = 0.0f;
    for (int c = 0; c < C; ++c) {
        float m = pl[c] + pr[c];
        m = (m > 0.0f) ? m : NEG_SLOPE * m;
        s += m * pa[c];
    }
    atomicMax(&maxbuf[(size_t)dst * heads + h], fenc(s));
}

// ---------------------------------------------------------------------------
// Edge pass 2: ex = exp(e - max[dst,h]); denom[dst,h] += ex;
//              acc[dst, h, :] += ex * xl[src, h, :]
// ---------------------------------------------------------------------------
__global__ void edge_acc(const float* __restrict__ xl, const float* __restrict__ xr,
                         const float* __restrict__ att, const int* __restrict__ ei,
                         int E, int N, int heads, int C,
                         const unsigned* __restrict__ maxbuf,
                         float* __restrict__ den, float* __restrict__ acc) {
    long long tid   = blockIdx.x * (long long)blockDim.x + threadIdx.x;
    long long total = (long long)(E + N) * heads;
    if (tid >= total) return;
    int h = (int)(tid % heads);
    int e = (int)(tid / heads);
    int src, dst;
    if (e < E) { src = ei[e]; dst = ei[E + e]; } else { src = dst = e - E; }
    int F = heads * C;
    const float* pl = xl + (size_t)src * F + h * C;
    const float* pr = xr + (size_t)dst * F + h * C;
    const float* pa = att + h * C;
    float s = 0.0f;
    for (int c = 0; c < C; ++c) {
        float m = pl[c] + pr[c];
        m = (m > 0.0f) ? m : NEG_SLOPE * m;
        s += m * pa[c];
    }
    float mx = fdec(maxbuf[(size_t)dst * heads + h]);
    float ex = __expf(s - mx);
    atomicAdd(&den[(size_t)dst * heads + h], ex);
    float* pd = acc + (size_t)dst * F + h * C;
    for (int c = 0; c < C; ++c) atomicAdd(&pd[c], ex * pl[c]);
}

// ---------------------------------------------------------------------------
// Finalize: out[n,f] = acc[n,f] / den[n,h] + bias[f]   (+ optional ELU)
// ---------------------------------------------------------------------------
__global__ void finalize_layer(const float* __restrict__ acc, const float* __restrict__ den,
                               const float* __restrict__ bias, float* __restrict__ out,
                               int N, int heads, int C, int do_elu) {
    int F = heads * C;
    long long tid = blockIdx.x * (long long)blockDim.x + threadIdx.x;
    if (tid >= (long long)N * F) return;
    int n = (int)(tid / F);
    int f = (int)(tid % F);
    int h = f / C;
    float v = acc[tid] / den[(size_t)n * heads + h] + bias[f];
    if (do_elu) v = (v > 0.0f) ? v : (__expf(v) - 1.0f);
    out[tid] = v;
}

// ---------------------------------------------------------------------------
// log_softmax over the 32 output channels per node
// ---------------------------------------------------------------------------
__global__ void log_softmax_k(const float* __restrict__ h, float* __restrict__ out, int N) {
    int n = blockIdx.x * blockDim.x + threadIdx.x;
    if (n >= N) return;
    const float* r = h + (size_t)n * DOUT;
    float mx = r[0];
    for (int c = 1; c < DOUT; ++c) mx = fmaxf(mx, r[c]);
    float s = 0.0f;
    for (int c = 0; c < DOUT; ++c) s += __expf(r[c] - mx);
    float lse = mx + __logf(s);
    float* o = out + (size_t)n * DOUT;
    for (int c = 0; c < DOUT; ++c) o[c] = r[c] - lse;
}

extern "C" void kernel_launch(void* const* d_in, const int* in_sizes, int n_in,
                              void* d_out, int out_size, void* d_ws, size_t ws_size,
                              hipStream_t stream) {
    const float* x    = (const float*)d_in[0];
    const int*   ei   = (const int*)d_in[1];
    /* d_in[2] = edge_attr (unused: edge_dim=None in reference) */
    const float* Wl1  = (const float*)d_in[3];
    const float* Wr1  = (const float*)d_in[4];
    const float* att1 = (const float*)d_in[5];
    const float* b1   = (const float*)d_in[6];
    const float* Wl2  = (const float*)d_in[7];
    const float* Wr2  = (const float*)d_in[8];
    const float* att2 = (const float*)d_in[9];
    const float* b2   = (const float*)d_in[10];

    const int N = in_sizes[0] / DIN;
    const int E = in_sizes[1] / 2;
    const size_t nN = (size_t)N;

    // Workspace layout (floats)
    float*    ws   = (float*)d_ws;
    float*    xl1  = ws;                     // N*128
    float*    xr1  = xl1 + nN * F1;          // N*128
    float*    acc1 = xr1 + nN * F1;          // N*128 (becomes h1 after finalize)
    float*    xl2  = acc1 + nN * F1;         // N*32
    float*    xr2  = xl2 + nN * DOUT;        // N*32
    float*    acc2 = xr2 + nN * DOUT;        // N*32
    float*    den1 = acc2 + nN * DOUT;       // N*8
    float*    den2 = den1 + nN * H1;         // N
    unsigned* max1 = (unsigned*)(den2 + nN); // N*8 (uint keys; 0 == minimum key)
    unsigned* max2 = max1 + nN * H1;         // N

    float* h_out  = (float*)d_out;           // N*32  (first tuple element)
    float* ls_out = h_out + nN * DOUT;       // N*32  (log_softmax)

    const int B = 256;
    auto cdiv = [](long long a, long long b) { return (unsigned)((a + b - 1) / b); };

    // Zero acc1..max2 (contiguous): N*(128+32+32+32+8+1+8+1) floats
    long long zn = nN * (long long)(F1 + 3 * DOUT + H1 + 1) + nN * (long long)(H1 + 1);
    zero_f32<<<cdiv(zn, B), B, 0, stream>>>(acc1, zn);

    // ---- Layer 1 -----------------------------------------------------------
    {
        long long waves = (long long)(N / 16) * (F1 / 16);
        unsigned g = cdiv(waves, 8);
        gemm_wmma_f32<<<g, B, 0, stream>>>(x, Wl1, xl1, N, DIN, F1);
        gemm_wmma_f32<<<g, B, 0, stream>>>(x, Wr1, xr1, N, DIN, F1);
    }
    {
        long long tot = (long long)(E + N) * H1;
        edge_max<<<cdiv(tot, B), B, 0, stream>>>(xl1, xr1, att1, ei, E, N, H1, C1, max1);
        edge_acc<<<cdiv(tot, B), B, 0, stream>>>(xl1, xr1, att1, ei, E, N, H1, C1,
                                                 max1, den1, acc1);
        long long nf = nN * F1;
        finalize_layer<<<cdiv(nf, B), B, 0, stream>>>(acc1, den1, b1, acc1, N, H1, C1, 1);
    }

    // ---- Layer 2 (heads=1, C=32) ------------------------------------------
    {
        long long waves = (long long)(N / 16) * (DOUT / 16);
        unsigned g = cdiv(waves, 8);
        gemm_wmma_f32<<<g, B, 0, stream>>>(acc1, Wl2, xl2, N, F1, DOUT);
        gemm_wmma_f32<<<g, B, 0, stream>>>(acc1, Wr2, xr2, N, F1, DOUT);
    }
    {
        long long tot = (long long)(E + N);
        edge_max<<<cdiv(tot, B), B, 0, stream>>>(xl2, xr2, att2, ei, E, N, 1, DOUT, max2);
        edge_acc<<<cdiv(tot, B), B, 0, stream>>>(xl2, xr2, att2, ei, E, N, 1, DOUT,
                                                 max2, den2, acc2);
        long long nf = nN * DOUT;
        finalize_layer<<<cdiv(nf, B), B, 0, stream>>>(acc2, den2, b2, h_out, N, 1, DOUT, 0);
    }

    // ---- log_softmax -------------------------------------------------------
    log_softmax_k<<<cdiv(N, B), B, 0, stream>>>(h_out, ls_out, N);
}